// FacetAttentionGNN_36240934043735
// MI455X (gfx1250) — compile-verified
//
#include <hip/hip_runtime.h>
#include <hip/hip_bf16.h>
#include <math.h>

typedef __attribute__((ext_vector_type(2))) float v2f;
typedef __attribute__((ext_vector_type(8))) float v8f;

#define NN 100000
#define BN_EPS 1e-5f

// ---------------- elementwise / utility kernels ----------------

__global__ void k_fill(float* __restrict__ p, float v, long n) {
  long i = (long)blockIdx.x * blockDim.x + threadIdx.x;
  if (i < n) p[i] = v;
}

__global__ void k_degree(const int* __restrict__ dst, float* __restrict__ deg, int E) {
  int e = blockIdx.x * blockDim.x + threadIdx.x;
  if (e < E) atomicAdd(&deg[dst[e]], 1.0f);
}

__global__ void k_rsqrt(float* __restrict__ d, int n) {
  int i = blockIdx.x * blockDim.x + threadIdx.x;
  if (i < n) d[i] = rsqrtf(d[i]);
}

// h (N,16) = x[:,3:6] @ W(3,16)   (K=3 not WMMA-friendly; trivial VALU kernel)
__global__ void k_gcn1_dense(const float* __restrict__ x, const float* __restrict__ W,
                             float* __restrict__ h, int N) {
  int i = blockIdx.x * blockDim.x + threadIdx.x;
  if (i >= N) return;
  float x0 = x[(size_t)i * 6 + 3], x1 = x[(size_t)i * 6 + 4], x2 = x[(size_t)i * 6 + 5];
#pragma unroll
  for (int c = 0; c < 16; ++c)
    h[(size_t)i * 16 + c] = x0 * W[c] + x1 * W[16 + c] + x2 * W[32 + c];
}

// agg[i] = h[i] * dis[row]^2   (self-loop contribution of GCN aggregation)
__global__ void k_selfloop(const float* __restrict__ h, const float* __restrict__ dis,
                           float* __restrict__ agg, long total, int C) {
  long i = (long)blockIdx.x * blockDim.x + threadIdx.x;
  if (i >= total) return;
  int row = (int)(i / C);
  float d = dis[row];
  agg[i] = h[i] * d * d;
}

// scatter-add: agg[dst] += h[src] * dis[src]*dis[dst]; thread per (edge, 4 channels)
template <int C>
__global__ void k_edge_scatter(const float* __restrict__ h, const int* __restrict__ src,
                               const int* __restrict__ dst, const float* __restrict__ dis,
                               float* __restrict__ agg, int E) {
  const int CG = C / 4;
  long tid = (long)blockIdx.x * blockDim.x + threadIdx.x;
  if (tid >= (long)E * CG) return;
  int e = (int)(tid / CG);
  int c0 = (int)(tid % CG) * 4;
  int s = src[e], d = dst[e];
  float w = dis[s] * dis[d];
  const float4 hv = *(const float4*)(h + (size_t)s * C + c0);
  float* ap = agg + (size_t)d * C + c0;
  atomicAdd(ap + 0, hv.x * w);
  atomicAdd(ap + 1, hv.y * w);
  atomicAdd(ap + 2, hv.z * w);
  atomicAdd(ap + 3, hv.w * w);
}

// ---------------- WMMA GEMM (fp32, V_WMMA_F32_16X16X4_F32) ----------------
// One wave -> 16 rows x FOUT cols.  A (16x4): lane<16 holds M=lane, K=k+{0,1};
// lane>=16 holds K=k+{2,3}.  B (4x16): same K split, N=lane&15.
// C/D (16x16): VGPR r -> M = r + 8*(lane>>4), N = lane&15.
// All optional epilogue/prologue features are compile-time to keep the inner
// loop branch-free: load -> pk_fma(scale,shift) -> max(relu) -> wmma.
template <int FIN, int FOUT, bool SCALE, bool SHIFT, bool RELU, bool RSCALE, bool BIAS>
__global__ __launch_bounds__(256) void k_gemm(
    const float* __restrict__ act, const float* __restrict__ Wm,
    const float* __restrict__ in_scale, const float* __restrict__ in_shift,
    const float* __restrict__ row_scale, const float* __restrict__ out_bias,
    float* __restrict__ out, int N) {
  const int CT = FOUT / 16;
  int lane = threadIdx.x & 31;
  int wv = threadIdx.x >> 5;
  int r0 = (blockIdx.x * 8 + wv) * 16;
  if (r0 >= N) return;  // whole-wave uniform exit; EXEC stays all-ones for WMMA
  int mr = lane & 15;
  int kk = (lane >> 4) << 1;
  int row = r0 + mr;
  float rs = 1.0f;
  if (RSCALE) rs = row_scale[row];
  v8f acc[CT];
#pragma unroll
  for (int ct = 0; ct < CT; ++ct)
#pragma unroll
    for (int j = 0; j < 8; ++j) acc[ct][j] = 0.0f;
  const float* ar = act + (size_t)row * FIN;
#pragma unroll
  for (int k = 0; k < FIN; k += 4) {
    int ka = k + kk;
    float ax = ar[ka];
    float ay = ar[ka + 1];
    if (SHIFT) {
      float s0 = SCALE ? in_scale[ka] : 1.0f;
      float s1 = SCALE ? in_scale[ka + 1] : 1.0f;
      ax = fmaf(ax, s0, in_shift[ka]);
      ay = fmaf(ay, s1, in_shift[ka + 1]);
    }
    if (RELU) { ax = fmaxf(ax, 0.0f); ay = fmaxf(ay, 0.0f); }
    if (RSCALE) { ax *= rs; ay *= rs; }
    v2f a; a.x = ax; a.y = ay;
#pragma unroll
    for (int ct = 0; ct < CT; ++ct) {
      int col = ct * 16 + mr;
      v2f b;
      b.x = Wm[(size_t)ka * FOUT + col];
      b.y = Wm[(size_t)(ka + 1) * FOUT + col];
      acc[ct] = __builtin_amdgcn_wmma_f32_16x16x4_f32(false, a, false, b,
                                                      (short)0, acc[ct], false, false);
    }
  }
  int half = (lane >> 4) << 3;
#pragma unroll
  for (int ct = 0; ct < CT; ++ct) {
    int col = ct * 16 + mr;
    float bv = BIAS ? out_bias[col] : 0.0f;
#pragma unroll
    for (int r = 0; r < 8; ++r)
      out[(size_t)(r0 + half + r) * FOUT + col] = acc[ct][r] + bv;
  }
}

// ---------------- BatchNorm statistics ----------------

__global__ __launch_bounds__(256) void k_colstats(const float* __restrict__ z, long total,
                                                  int C, float* __restrict__ stat) {
  __shared__ float ss[256], sq[256];
  int t = threadIdx.x;
  float s = 0.f, q = 0.f;
  for (long i = (long)blockIdx.x * 256 + t; i < total; i += (long)gridDim.x * 256) {
    float v = z[i];
    s += v; q += v * v;
  }
  ss[t] = s; sq[t] = q;
  __syncthreads();
  for (int off = 128; off >= C; off >>= 1) {
    if (t < off) { ss[t] += ss[t + off]; sq[t] += sq[t + off]; }
    __syncthreads();
  }
  if (t < C) {
    atomicAdd(&stat[t], ss[t]);
    atomicAdd(&stat[C + t], sq[t]);
  }
}

__global__ void k_bn_finalize(const float* __restrict__ stat, const float* __restrict__ g,
                              const float* __restrict__ be, int C, float invN,
                              float* __restrict__ sc, float* __restrict__ sh) {
  int c = threadIdx.x;
  if (c >= C) return;
  float mean = stat[c] * invN;
  float var = stat[C + c] * invN - mean * mean;
  float s = g[c] * rsqrtf(var + BN_EPS);
  sc[c] = s;
  sh[c] = be[c] - mean * s;
}

// column max of relu(z*sc+sh); result >= 0 so uint-bit atomicMax is order-correct
__global__ __launch_bounds__(256) void k_colmax(const float* __restrict__ z, long total, int C,
                                                const float* __restrict__ sc,
                                                const float* __restrict__ sh,
                                                float* __restrict__ outmax) {
  __shared__ float sm[256];
  int t = threadIdx.x;
  int c = t % C;
  float s = sc[c], h = sh[c];
  float m = 0.f;
  for (long i = (long)blockIdx.x * 256 + t; i < total; i += (long)gridDim.x * 256)
    m = fmaxf(m, fmaxf(fmaf(z[i], s, h), 0.f));
  sm[t] = m;
  __syncthreads();
  for (int off = 128; off >= C; off >>= 1) {
    if (t < off) sm[t] = fmaxf(sm[t], sm[t + off]);
    __syncthreads();
  }
  if (t < C) atomicMax((unsigned int*)&outmax[t], __float_as_uint(sm[t]));
}

// out = act(z*sc+sh); mode 0=none 1=relu 2=sigmoid
__global__ void k_affine_act(const float* __restrict__ z, const float* __restrict__ sc,
                             const float* __restrict__ sh, float* __restrict__ out,
                             long total, int C, int mode) {
  long i = (long)blockIdx.x * blockDim.x + threadIdx.x;
  if (i >= total) return;
  int c = (int)(i % C);
  float s = sc ? sc[c] : 1.0f;
  float v = fmaf(z[i], s, sh[c]);
  if (mode == 1) v = fmaxf(v, 0.f);
  else if (mode == 2) v = 1.0f / (1.0f + expf(-v));
  out[i] = v;
}

__global__ void k_sub(const float* __restrict__ a, const float* __restrict__ b,
                      float* __restrict__ o, long n) {
  long i = (long)blockIdx.x * blockDim.x + threadIdx.x;
  if (i < n) o[i] = a[i] - b[i];
}

// beff[c] = mf0b[c] + sum_k fg[k] * W160[(32+k),c]  (fg is row-constant -> fold to bias)
__global__ void k_fold_bias(const float* __restrict__ fg, const float* __restrict__ W160,
                            const float* __restrict__ b, float* __restrict__ beff) {
  int c = threadIdx.x;
  if (c >= 128) return;
  float acc = b[c];
  for (int k = 0; k < 128; ++k) acc = fmaf(fg[k], W160[(size_t)(32 + k) * 128 + c], acc);
  beff[c] = acc;
}

// ch = [max(fd2,axis=1), mean(fd2,axis=1)]; z = ch @ ma0W(2,32) + ma0b
__global__ void k_ch_ma0(const float* __restrict__ fd, const float* __restrict__ W,
                         const float* __restrict__ b, float* __restrict__ z, int N) {
  int i = blockIdx.x * blockDim.x + threadIdx.x;
  if (i >= N) return;
  const float* r = fd + (size_t)i * 32;
  float mx = -3.4e38f, sum = 0.f;
#pragma unroll
  for (int c = 0; c < 32; ++c) { float v = r[c]; mx = fmaxf(mx, v); sum += v; }
  float mean = sum * (1.0f / 32.0f);
#pragma unroll
  for (int c = 0; c < 32; ++c)
    z[(size_t)i * 32 + c] = fmaf(mx, W[c], fmaf(mean, W[32 + c], b[c]));
}

// ma2: z = relu(bn(zin)) @ W(32,1) + b
__global__ void k_lin32to1(const float* __restrict__ zin, const float* __restrict__ sc,
                           const float* __restrict__ sh, const float* __restrict__ W,
                           const float* __restrict__ b, float* __restrict__ zout, int N) {
  int i = blockIdx.x * blockDim.x + threadIdx.x;
  if (i >= N) return;
  const float* r = zin + (size_t)i * 32;
  float acc = b[0];
#pragma unroll
  for (int c = 0; c < 32; ++c) {
    float v = fmaxf(fmaf(r[c], sc[c], sh[c]), 0.f);
    acc = fmaf(v, W[c], acc);
  }
  zout[i] = acc;
}

// fc2 (32->3) + row L2 normalize
__global__ void k_fc2_norm(const float* __restrict__ f1, const float* __restrict__ W,
                           const float* __restrict__ b, float* __restrict__ out, int N) {
  int i = blockIdx.x * blockDim.x + threadIdx.x;
  if (i >= N) return;
  const float* r = f1 + (size_t)i * 32;
  float o0 = b[0], o1 = b[1], o2 = b[2];
#pragma unroll
  for (int c = 0; c < 32; ++c) {
    float v = r[c];
    o0 = fmaf(v, W[c * 3 + 0], o0);
    o1 = fmaf(v, W[c * 3 + 1], o1);
    o2 = fmaf(v, W[c * 3 + 2], o2);
  }
  float inv = 1.0f / fmaxf(sqrtf(o0 * o0 + o1 * o1 + o2 * o2), 1e-12f);
  out[(size_t)i * 3 + 0] = o0 * inv;
  out[(size_t)i * 3 + 1] = o1 * inv;
  out[(size_t)i * 3 + 2] = o2 * inv;
}

// ---------------- host orchestration ----------------

enum {
  GCN1W = 0, GCN1B, GCN3W, GCN3B, GCN5W, GCN5B, GCN6W, GCN6B,
  MG0W, MG0B, MG1W, MG1B, MF0W, MF0B, MF1W, MF1B,
  MD0W, MD0B, MD1W, MD1B, MA0W, MA0B, MA1W, MA1B, MA2W, MA2B,
  FC1W, FC1B, FC2W, FC2B,
  MG0G, MG0BE, MG1G, MG1BE, MF0G, MF0BE, MF1G, MF1BE,
  MD0G, MD0BE, MD1G, MD1BE, MA0G, MA0BE, MA1G, MA1BE, MA2G, MA2BE,
  NPARAM
};

static inline int cdiv(long a, long b) { return (int)((a + b - 1) / b); }

extern "C" void kernel_launch(void* const* d_in, const int* in_sizes, int n_in,
                              void* d_out, int out_size, void* d_ws, size_t ws_size,
                              hipStream_t stream) {
  const int N = NN;
  const int TB = 256;
  const float* x = (const float*)d_in[0];
  const int* ei = (const int*)d_in[1];
  const int E = in_sizes[1] / 2;
  const int* src = ei;
  const int* dstp = ei + E;
  const float* P[NPARAM];
  for (int i = 0; i < NPARAM; ++i) P[i] = (i + 2 < n_in) ? (const float*)d_in[2 + i] : nullptr;

  // workspace layout
  float* Wp = (float*)d_ws;
  float* dis = Wp;                   // N
  float* sm = Wp + N;                // 8192 small pool
  float* fg = sm;                    // 128
  float* beff = sm + 128;            // 128
  float* stat = sm + 256;            // 256
  float* bn = sm + 512;
  float *sc_mg0 = bn + 0,   *sh_mg0 = bn + 64;
  float *sc_mg1 = bn + 128, *sh_mg1 = bn + 256;
  float *sc_mf0 = bn + 384, *sh_mf0 = bn + 512;
  float *sc_mf1 = bn + 640, *sh_mf1 = bn + 672;
  float *sc_md0 = bn + 704, *sh_md0 = bn + 736;
  float *sc_md1 = bn + 768, *sh_md1 = bn + 800;
  float *sc_ma0 = bn + 832, *sh_ma0 = bn + 864;
  float *sc_ma1 = bn + 896, *sh_ma1 = bn + 928;
  float *sc_ma2 = bn + 960, *sh_ma2 = bn + 962;
  float* big = Wp + N + 8192;
  float* T16A = big;  big += (size_t)N * 16;
  float* T16B = big;  big += (size_t)N * 16;
  float* T32A = big;  big += (size_t)N * 32;  // FEAT
  float* T32B = big;  big += (size_t)N * 32;
  float* T32C = big;  big += (size_t)N * 32;  // FA
  float* T32D = big;  big += (size_t)N * 32;
  float* T64  = big;  big += (size_t)N * 64;
  float* T128A = big; big += (size_t)N * 128;
  float* T128B = big; big += (size_t)N * 128;
  float* T1 = big;

  const int gGemm = cdiv(N, 16 * 8);
  const float invN = 1.0f / (float)N;

#define STATS(zbuf, C, g, be, sc, sh)                                        \
  do {                                                                       \
    k_fill<<<1, 256, 0, stream>>>(stat, 0.f, 2 * (C));                       \
    k_colstats<<<256, 256, 0, stream>>>(zbuf, (long)N * (C), (C), stat);     \
    k_bn_finalize<<<1, 128, 0, stream>>>(stat, g, be, (C), invN, sc, sh);    \
  } while (0)

  // degree / rsqrt (deg init 1.0 covers self-loops)
  k_fill<<<cdiv(N, TB), TB, 0, stream>>>(dis, 1.0f, N);
  k_degree<<<cdiv(E, TB), TB, 0, stream>>>(dstp, dis, E);
  k_rsqrt<<<cdiv(N, TB), TB, 0, stream>>>(dis, N);

  // gcn1: dense 3->16, aggregate
  k_gcn1_dense<<<cdiv(N, TB), TB, 0, stream>>>(x, P[GCN1W], T16A, N);
  k_selfloop<<<cdiv((long)N * 16, TB), TB, 0, stream>>>(T16A, dis, T16B, (long)N * 16, 16);
  k_edge_scatter<16><<<cdiv((long)E * 4, TB), TB, 0, stream>>>(T16A, src, dstp, dis, T16B, E);

  // gcn3: (relu(agg1 + b1)) @ W3 fused, then aggregate, feat = relu(agg3 + b3)
  k_gemm<16, 32, false, true, true, false, false><<<gGemm, TB, 0, stream>>>(
      T16B, P[GCN3W], nullptr, P[GCN1B], nullptr, nullptr, T32A, N);
  k_selfloop<<<cdiv((long)N * 32, TB), TB, 0, stream>>>(T32A, dis, T32B, (long)N * 32, 32);
  k_edge_scatter<32><<<cdiv((long)E * 8, TB), TB, 0, stream>>>(T32A, src, dstp, dis, T32B, E);
  k_affine_act<<<cdiv((long)N * 32, TB), TB, 0, stream>>>(T32B, nullptr, P[GCN3B], T32A,
                                                          (long)N * 32, 32, 1);  // FEAT

  // mg0 / mg1 / global max -> fg
  k_gemm<32, 64, false, false, false, false, true><<<gGemm, TB, 0, stream>>>(
      T32A, P[MG0W], nullptr, nullptr, nullptr, P[MG0B], T64, N);
  STATS(T64, 64, P[MG0G], P[MG0BE], sc_mg0, sh_mg0);
  k_gemm<64, 128, true, true, true, false, true><<<gGemm, TB, 0, stream>>>(
      T64, P[MG1W], sc_mg0, sh_mg0, nullptr, P[MG1B], T128A, N);
  STATS(T128A, 128, P[MG1G], P[MG1BE], sc_mg1, sh_mg1);
  k_fill<<<1, 128, 0, stream>>>(fg, 0.f, 128);
  k_colmax<<<256, 256, 0, stream>>>(T128A, (long)N * 128, 128, sc_mg1, sh_mg1, fg);

  // mf0: fold fg into bias, 32->128; mf1: 128->32; FA = relu(bn(z))
  k_fold_bias<<<1, 128, 0, stream>>>(fg, P[MF0W], P[MF0B], beff);
  k_gemm<32, 128, false, false, false, false, true><<<gGemm, TB, 0, stream>>>(
      T32A, P[MF0W], nullptr, nullptr, nullptr, beff, T128A, N);
  STATS(T128A, 128, P[MF0G], P[MF0BE], sc_mf0, sh_mf0);
  k_gemm<128, 32, true, true, true, false, true><<<gGemm, TB, 0, stream>>>(
      T128A, P[MF1W], sc_mf0, sh_mf0, nullptr, P[MF1B], T32B, N);
  STATS(T32B, 32, P[MF1G], P[MF1BE], sc_mf1, sh_mf1);
  k_affine_act<<<cdiv((long)N * 32, TB), TB, 0, stream>>>(T32B, sc_mf1, sh_mf1, T32C,
                                                          (long)N * 32, 32, 1);  // FA

  // fd = feat - fa; md0; md1; fd2
  k_sub<<<cdiv((long)N * 32, TB), TB, 0, stream>>>(T32A, T32C, T32B, (long)N * 32);
  k_gemm<32, 32, false, false, false, false, true><<<gGemm, TB, 0, stream>>>(
      T32B, P[MD0W], nullptr, nullptr, nullptr, P[MD0B], T32D, N);
  STATS(T32D, 32, P[MD0G], P[MD0BE], sc_md0, sh_md0);
  k_gemm<32, 32, true, true, true, false, true><<<gGemm, TB, 0, stream>>>(
      T32D, P[MD1W], sc_md0, sh_md0, nullptr, P[MD1B], T32B, N);
  STATS(T32B, 32, P[MD1G], P[MD1BE], sc_md1, sh_md1);
  k_affine_act<<<cdiv((long)N * 32, TB), TB, 0, stream>>>(T32B, sc_md1, sh_md1, T32D,
                                                          (long)N * 32, 32, 1);  // fd2

  // attention: ch -> ma0 -> ma1 -> ma2 -> sigmoid
  k_ch_ma0<<<cdiv(N, TB), TB, 0, stream>>>(T32D, P[MA0W], P[MA0B], T32B, N);
  STATS(T32B, 32, P[MA0G], P[MA0BE], sc_ma0, sh_ma0);
  k_gemm<32, 32, true, true, true, false, true><<<gGemm, TB, 0, stream>>>(
      T32B, P[MA1W], sc_ma0, sh_ma0, nullptr, P[MA1B], T32D, N);
  STATS(T32D, 32, P[MA1G], P[MA1BE], sc_ma1, sh_ma1);
  k_lin32to1<<<cdiv(N, TB), TB, 0, stream>>>(T32D, sc_ma1, sh_ma1, P[MA2W], P[MA2B], T1, N);
  STATS(T1, 1, P[MA2G], P[MA2BE], sc_ma2, sh_ma2);
  k_affine_act<<<cdiv(N, TB), TB, 0, stream>>>(T1, sc_ma2, sh_ma2, T1, N, 1, 2);  // a

  // gcn5: (fa * a) @ W5, aggregate
  k_gemm<32, 32, false, false, false, true, false><<<gGemm, TB, 0, stream>>>(
      T32C, P[GCN5W], nullptr, nullptr, T1, nullptr, T32B, N);
  k_selfloop<<<cdiv((long)N * 32, TB), TB, 0, stream>>>(T32B, dis, T32D, (long)N * 32, 32);
  k_edge_scatter<32><<<cdiv((long)E * 8, TB), TB, 0, stream>>>(T32B, src, dstp, dis, T32D, E);

  // gcn6: (agg5 + b5) @ W6, aggregate
  k_gemm<32, 128, false, true, false, false, false><<<gGemm, TB, 0, stream>>>(
      T32D, P[GCN6W], nullptr, P[GCN5B], nullptr, nullptr, T128A, N);
  k_selfloop<<<cdiv((long)N * 128, TB), TB, 0, stream>>>(T128A, dis, T128B, (long)N * 128, 128);
  k_edge_scatter<128><<<cdiv((long)E * 32, TB), TB, 0, stream>>>(T128A, src, dstp, dis, T128B, E);

  // fc1: (agg6 + b6) @ fc1W + fc1b; fc2 + normalize -> d_out
  k_gemm<128, 32, false, true, false, false, true><<<gGemm, TB, 0, stream>>>(
      T128B, P[FC1W], nullptr, P[GCN6B], nullptr, P[FC1B], T32B, N);
  k_fc2_norm<<<cdiv(N, TB), TB, 0, stream>>>(T32B, P[FC2W], P[FC2B], (float*)d_out, N);

#undef STATS
  (void)ws_size; (void)out_size;
}